// TaskAlignedAssigner_5763846111777
// MI455X (gfx1250) — compile-verified
//
#include <hip/hip_runtime.h>
#include <math.h>

#define BS 16
#define NA 8400
#define NM 64
#define NC 80
#define NTOPK 13
#define ATILES 525          // NA/16
#define CIOU_EPS 1e-7f
#define INV_PI2_4 0.4052847345693511f

// output layout (floats, concatenated in return order)
#define O_LBL   0u
#define O_BOX   134400u
#define O_SCR   672000u
#define O_FG    11424000u
#define O_TGI   11558400u

typedef __attribute__((ext_vector_type(2))) float v2f;
typedef __attribute__((ext_vector_type(8))) float v8f;

// ---------------------------------------------------------------------------
// K1: bbox_scores via one-hot f32 WMMA (exact gather) fused with CIoU metric.
// grid (66, BS), 256 thr = 8 waves; each wave owns one 16-anchor tile and all
// 64 GTs (4 WMMA accumulator tiles of 16x16).
// ---------------------------------------------------------------------------
__global__ __launch_bounds__(256) void k1_metric(
    const float* __restrict__ pd_scores, const float* __restrict__ pd_bboxes,
    const float* __restrict__ anc, const int* __restrict__ gt_labels,
    const float* __restrict__ gt_bboxes, const float* __restrict__ mask_gt,
    float* __restrict__ metric, float* __restrict__ ovl,
    unsigned char* __restrict__ mgtA)
{
  __shared__ float gx1[NM], gy1[NM], gx2[NM], gy2[NM], gat[NM], gmk[NM];
  __shared__ int glb[NM];
  const int tid = threadIdx.x;
  const int b = blockIdx.y;
  if (tid < NM) {
    const int gi = b * NM + tid;
    float x1 = gt_bboxes[gi*4+0], y1 = gt_bboxes[gi*4+1];
    float x2 = gt_bboxes[gi*4+2], y2 = gt_bboxes[gi*4+3];
    gx1[tid]=x1; gy1[tid]=y1; gx2[tid]=x2; gy2[tid]=y2;
    gat[tid] = atanf((x2-x1) / (y2-y1 + CIOU_EPS));    // atan(w1/h1), hoisted
    gmk[tid] = mask_gt[gi];
    int l = gt_labels[gi];
    l = l < 0 ? 0 : (l > NC-1 ? NC-1 : l);
    glb[tid] = l;
  }
  __syncthreads();

  const int wid  = tid >> 5;
  const int lane = tid & 31;
  const int tile = blockIdx.x * 8 + wid;
  if (tile >= ATILES) return;
  const int hf = lane >> 4;      // lane half selects K pairs per ISA layout
  const int lq = lane & 15;
  const int a  = tile * 16 + lq; // this lane's anchor (N index of B/D)

  const float* Srow = pd_scores + ((size_t)b * NA + a) * NC; // scores[a, :]
  __builtin_prefetch(Srow, 0, 0); // global_prefetch_b8: pull 320B row early

  v8f acc[4];
  #pragma unroll
  for (int t = 0; t < 4; ++t) acc[t] = (v8f){0,0,0,0,0,0,0,0};
  const int cm[4] = { glb[lq], glb[16+lq], glb[32+lq], glb[48+lq] };

  for (int k0 = 0; k0 < NC; k0 += 4) {
    const int kb = k0 + 2*hf;              // VGPR0 -> K=kb, VGPR1 -> K=kb+1
    v2f Bv;                                // B[k][n] = scores[a=n, c=k]
    Bv.x = Srow[kb];
    Bv.y = Srow[kb+1];
    #pragma unroll
    for (int t = 0; t < 4; ++t) {
      v2f Av;                              // one-hot A[m][k], m = 16t+lq
      Av.x = (cm[t] == kb    ) ? 1.f : 0.f;
      Av.y = (cm[t] == kb + 1) ? 1.f : 0.f;
      acc[t] = __builtin_amdgcn_wmma_f32_16x16x4_f32(
          false, Av, false, Bv, (short)0, acc[t], false, false);
    }
  }

  // per-anchor (lane-constant) box terms, hoisted out of the 32-pair loop
  const float4 pb = ((const float4*)pd_bboxes)[(size_t)b * NA + a];
  const float ax = anc[2*a], ay = anc[2*a+1];
  const float w2 = pb.z - pb.x, h2 = pb.w - pb.y + CIOU_EPS;
  const float at2 = atanf(w2 / h2);
  const float area2 = w2 * h2;

  #pragma unroll
  for (int t = 0; t < 4; ++t) {
    #pragma unroll
    for (int v = 0; v < 8; ++v) {
      const int m = 16*t + v + 8*hf;       // D layout: VGPR v -> row M
      const float x1 = gx1[m], y1 = gy1[m], x2 = gx2[m], y2 = gy2[m];
      const float gm = gmk[m];             // unconditional LDS read
      const float w1 = x2 - x1, h1 = y2 - y1 + CIOU_EPS;
      // mask_in_gts & mask_gt — non-short-circuit: stays as VCC mask ops,
      // no EXEC-manipulating branch inside the 32-pair loop.
      const float dmin = fminf(fminf(ax - x1, ay - y1), fminf(x2 - ax, y2 - ay));
      const int mgt = (int)(dmin > 1e-9f) & (int)(gm > 0.f);
      // CIoU (gt = box1, pd = box2), f32, reference op order
      float iw = fminf(x2, pb.z) - fmaxf(x1, pb.x); iw = fmaxf(iw, 0.f);
      float ih = fminf(y2, pb.w) - fmaxf(y1, pb.y); ih = fmaxf(ih, 0.f);
      const float inter = iw * ih;
      const float uni = w1*h1 + area2 - inter + CIOU_EPS;
      const float iou = inter / uni;
      const float cw = fmaxf(x2, pb.z) - fminf(x1, pb.x);
      const float ch = fmaxf(y2, pb.w) - fminf(y1, pb.y);
      const float c2 = cw*cw + ch*ch + CIOU_EPS;
      const float dx = pb.x + pb.z - x1 - x2;
      const float dy = pb.y + pb.w - y1 - y2;
      const float rho2 = (dx*dx + dy*dy) * 0.25f;
      const float da = at2 - gat[m];
      const float vv = INV_PI2_4 * da * da;
      const float alpha = vv / (vv - iou + (1.f + CIOU_EPS));
      float ciou = iou - (rho2 / c2 + vv * alpha);
      ciou = fmaxf(ciou, 0.f);
      const float ov = (ciou < 0.5f) ? ciou * ciou : sqrtf(ciou);
      const float score = acc[t][v];       // WMMA-gathered pd_scores[a, label_m]
      const float ov2 = ov * ov;
      const float met = mgt ? score * (ov2 * ov2 * ov2) : 0.f;  // s^1 * o^6
      const size_t ix = ((size_t)(b*NM + m)) * NA + a;
      metric[ix] = met;
      ovl[ix]    = mgt ? ov : 0.f;
      mgtA[ix]   = (unsigned char)mgt;
    }
  }
}

// ---------------------------------------------------------------------------
// K2: exact per-(b,m) top-13 with lowest-index tie-break (matches lax.top_k),
// row resident in LDS; mask_pos = selected && mgt; mask_gt==0 -> zero row.
// ---------------------------------------------------------------------------
__global__ __launch_bounds__(256) void k2_topk(
    const float* __restrict__ metric, const unsigned char* __restrict__ mgtA,
    const float* __restrict__ mask_gt, unsigned char* __restrict__ mpos)
{
  __shared__ float lm[NA];
  __shared__ unsigned char sel[NA];
  __shared__ float rv[256];
  __shared__ int ri[256];
  const int bm = blockIdx.x;
  const int tid = threadIdx.x;
  const size_t row = (size_t)bm * NA;
  if (mask_gt[bm] <= 0.f) {
    for (int a = tid; a < NA; a += 256) mpos[row + a] = 0;
    return;
  }
  for (int a = tid; a < NA; a += 256) { lm[a] = metric[row + a]; sel[a] = 0; }
  __syncthreads();
  for (int r = 0; r < NTOPK; ++r) {
    float bv = -1.f; int bi = NA;
    for (int a = tid; a < NA; a += 256) {      // ascending -> keeps lowest idx
      const float v = lm[a];
      if (v > bv) { bv = v; bi = a; }
    }
    rv[tid] = bv; ri[tid] = bi;
    __syncthreads();
    for (int s = 128; s > 0; s >>= 1) {
      if (tid < s) {
        const float v2 = rv[tid+s]; const int i2 = ri[tid+s];
        if (v2 > rv[tid] || (v2 == rv[tid] && i2 < ri[tid])) { rv[tid]=v2; ri[tid]=i2; }
      }
      __syncthreads();
    }
    if (tid == 0) { sel[ri[0]] = 1; lm[ri[0]] = -1.f; }
    __syncthreads();
  }
  for (int a = tid; a < NA; a += 256)
    mpos[row + a] = (unsigned char)((int)(sel[a] != 0) & (int)(mgtA[row + a] != 0));
}

// ---------------------------------------------------------------------------
// K3: per-anchor multi-assignment fix + labels/bboxes/fg/gt_idx outputs.
// ---------------------------------------------------------------------------
__global__ __launch_bounds__(256) void k3_resolve(
    unsigned char* __restrict__ mpos, const float* __restrict__ ovl,
    const int* __restrict__ gt_labels, const float* __restrict__ gt_bboxes,
    float* __restrict__ out)
{
  const int a = blockIdx.x * 256 + threadIdx.x;
  const int b = blockIdx.y;
  if (a >= NA) return;
  const size_t col = (size_t)b * NM * NA + a;
  int fg = 0, firstm = -1;
  for (int m = 0; m < NM; ++m)
    if (mpos[col + (size_t)m * NA]) { fg++; if (firstm < 0) firstm = m; }
  int tgt; float fgm;
  if (fg > 1) {
    float bv = -1.f; int bmx = 0;
    for (int m = 0; m < NM; ++m) {           // first-max == jnp.argmax
      const float v = ovl[col + (size_t)m * NA];
      if (v > bv) { bv = v; bmx = m; }
    }
    for (int m = 0; m < NM; ++m)
      mpos[col + (size_t)m * NA] = (m == bmx) ? 1 : 0;
    tgt = bmx; fgm = 1.f;
  } else {
    tgt = (firstm < 0) ? 0 : firstm;
    fgm = (float)fg;
  }
  const int gi = b * NM + tgt;
  int lbl = gt_labels[gi]; if (lbl < 0) lbl = 0;
  const size_t oi = (size_t)b * NA + a;
  out[O_LBL + oi] = (float)lbl;
  const float4 gb = ((const float4*)gt_bboxes)[gi];
  float* ob = out + O_BOX + oi * 4;
  ob[0] = gb.x; ob[1] = gb.y; ob[2] = gb.z; ob[3] = gb.w;
  out[O_FG  + oi] = (fgm > 0.f) ? 1.f : 0.f;
  out[O_TGI + oi] = (float)tgt;
}

// ---------------------------------------------------------------------------
// K4a: per-(b,m) pos_align / pos_overlaps (maxima of masked rows, post-fix).
// ---------------------------------------------------------------------------
__global__ __launch_bounds__(256) void k4a_rowmax(
    const float* __restrict__ metric, const float* __restrict__ ovl,
    const unsigned char* __restrict__ mpos,
    float* __restrict__ palign, float* __restrict__ povl)
{
  __shared__ float ra[256], ro[256];
  const int bm = blockIdx.x, tid = threadIdx.x;
  const size_t row = (size_t)bm * NA;
  float ma = 0.f, mo = 0.f;
  for (int a = tid; a < NA; a += 256) {
    const float msk = mpos[row + a] ? 1.f : 0.f;   // branch-free masked max
    ma = fmaxf(ma, metric[row + a] * msk);
    mo = fmaxf(mo, ovl[row + a] * msk);
  }
  ra[tid] = ma; ro[tid] = mo;
  __syncthreads();
  for (int s = 128; s > 0; s >>= 1) {
    if (tid < s) { ra[tid] = fmaxf(ra[tid], ra[tid+s]); ro[tid] = fmaxf(ro[tid], ro[tid+s]); }
    __syncthreads();
  }
  if (tid == 0) { palign[bm] = ra[0]; povl[bm] = ro[0]; }
}

// ---------------------------------------------------------------------------
// K4b: per-anchor norm + one-hot target_scores (coalesced 80-float rows).
// ---------------------------------------------------------------------------
__global__ __launch_bounds__(256) void k4b_scores(
    const float* __restrict__ metric, const unsigned char* __restrict__ mpos,
    const float* __restrict__ palign, const float* __restrict__ povl,
    float* __restrict__ out)
{
  __shared__ float pa[NM], po[NM];
  const int tid = threadIdx.x, b = blockIdx.y;
  if (tid < NM) { pa[tid] = palign[b*NM + tid]; po[tid] = povl[b*NM + tid]; }
  __syncthreads();
  const int a = blockIdx.x * 256 + tid;
  if (a >= NA) return;
  const size_t col = (size_t)b * NM * NA + a;
  float norm = 0.f;
  for (int m = 0; m < NM; ++m) {
    const float msk = mpos[col + (size_t)m * NA] ? 1.f : 0.f;
    const float t = metric[col + (size_t)m * NA] * msk * po[m] / (pa[m] + 1e-9f);
    norm = fmaxf(norm, t);
  }
  const size_t oi = (size_t)b * NA + a;
  const float fg = out[O_FG + oi];
  const int lbl = (int)out[O_LBL + oi];
  const float sval = (fg > 0.f) ? norm : 0.f;      // hoisted out of the c-loop
  float* os = out + O_SCR + oi * NC;
  for (int c = 0; c < NC; ++c)
    os[c] = (c == lbl) ? sval : 0.f;
}

// ---------------------------------------------------------------------------
extern "C" void kernel_launch(void* const* d_in, const int* in_sizes, int n_in,
                              void* d_out, int out_size, void* d_ws, size_t ws_size,
                              hipStream_t stream) {
  const float* pd_scores = (const float*)d_in[0];
  const float* pd_bboxes = (const float*)d_in[1];
  const float* anc       = (const float*)d_in[2];
  const int*   gt_labels = (const int*)d_in[3];
  const float* gt_bboxes = (const float*)d_in[4];
  const float* mask_gt   = (const float*)d_in[5];
  float* out = (float*)d_out;

  char* ws = (char*)d_ws;
  const size_t NPAIR = (size_t)BS * NM * NA;         // 8,601,600
  float* metric        = (float*)(ws);
  float* ovl           = (float*)(ws + NPAIR * 4);
  unsigned char* mgtA  = (unsigned char*)(ws + NPAIR * 8);
  unsigned char* mpos  = (unsigned char*)(ws + NPAIR * 9);
  float* palign        = (float*)(ws + NPAIR * 10);
  float* povl          = (float*)(ws + NPAIR * 10 + (size_t)BS * NM * 4);

  k1_metric <<<dim3(66, BS), 256, 0, stream>>>(pd_scores, pd_bboxes, anc,
                                               gt_labels, gt_bboxes, mask_gt,
                                               metric, ovl, mgtA);
  k2_topk   <<<BS * NM, 256, 0, stream>>>(metric, mgtA, mask_gt, mpos);
  k3_resolve<<<dim3(33, BS), 256, 0, stream>>>(mpos, ovl, gt_labels, gt_bboxes, out);
  k4a_rowmax<<<BS * NM, 256, 0, stream>>>(metric, ovl, mpos, palign, povl);
  k4b_scores<<<dim3(33, BS), 256, 0, stream>>>(metric, mpos, palign, povl, out);
}